// GCNClauseActionEmbedder_78950088835420
// MI455X (gfx1250) — compile-verified
//
#include <hip/hip_runtime.h>
#include <math.h>

// ---- problem constants (per reference setup_inputs) ----
#define BB   32
#define NN   2048
#define BN   (BB*NN)          // 65536 nodes
#define DD   64
#define SS   16
#define LL   3
#define NT   8                // edge types
#define SELF_LOOP 1
#define MAXDEP 31
#define NEL  ((size_t)BN*DD)  // 4,194,304 floats per node-feature buffer

typedef __attribute__((ext_vector_type(2))) float v2f;
typedef __attribute__((ext_vector_type(8))) float v8f;

// ---------------- utility kernels ----------------
__global__ void k_zero_f32(float* __restrict__ p, long n) {
    long i = (long)blockIdx.x * blockDim.x + threadIdx.x;
    if (i < n) p[i] = 0.0f;
}

// depths[flat_src] = max(dep) over self-loop edges
__global__ void k_scatter_depth(const int* __restrict__ adj, int E, int* __restrict__ depths) {
    int e = blockIdx.x * blockDim.x + threadIdx.x;
    if (e >= E) return;
    const int* t = adj + (size_t)e * 5;
    if (t[2] != SELF_LOOP) return;
    atomicMax(&depths[t[0] * NN + t[1]], t[4]);
}

// h[node] = concat(type_table[batch_nodes[node]], depth_table[min(depth,31)])
__global__ void k_embed(const int* __restrict__ bn, const int* __restrict__ depths,
                        const float* __restrict__ ttab, const float* __restrict__ dtab,
                        float* __restrict__ h) {
    long i = (long)blockIdx.x * blockDim.x + threadIdx.x;
    if (i >= (long)NEL) return;
    int node = (int)(i >> 6);
    int d = (int)(i & 63);
    float v;
    if (d < 32) {
        v = ttab[bn[node] * 32 + d];
    } else {
        int dep = depths[node]; if (dep > MAXDEP) dep = MAXDEP;
        v = dtab[dep * 32 + (d - 32)];
    }
    h[i] = v;
}

// typed segment-sum: SEG[dst] += h[src] for edges with et == t
// 16 threads per edge, 4 dims each (float4 gather + 4 native fp32 atomic adds)
__global__ void k_scatter_type(const int* __restrict__ adj, int E, int t,
                               const float* __restrict__ h, float* __restrict__ seg) {
    long tid = (long)blockIdx.x * blockDim.x + threadIdx.x;
    long e = tid >> 4;
    if (e >= E) return;
    const int* tp = adj + e * 5;
    if (tp[2] != t) return;
    int q = (int)(tid & 15);
    long fsrc = (long)tp[0] * NN + tp[1];
    long fdst = (long)tp[0] * NN + tp[3];
    const float4 v = *(const float4*)(h + fsrc * DD + q * 4);
    float* dp = seg + fdst * DD + q * 4;
    unsafeAtomicAdd(dp + 0, v.x);
    unsafeAtomicAdd(dp + 1, v.y);
    unsafeAtomicAdd(dp + 2, v.z);
    unsafeAtomicAdd(dp + 3, v.w);
}

// ---------------- WMMA fp32 matmul: Y (+)= X(BN x 64) @ W(64 x 64) [+bias][relu] ----------------
// block = 256 threads = 8 wave32; wave w handles rows [blk*128 + w*16, +16).
// V_WMMA_F32_16X16X4_F32: A 16x4, B 4x16, C/D 16x16 fp32.
// K-outer / N-tile-inner: each A fragment loaded once, shared by 4 WMMA chains.
__global__ void k_matmul64(const float* __restrict__ X, const float* __restrict__ W,
                           const float* __restrict__ bias, float* __restrict__ Y,
                           int accum, int has_bias, int do_relu) {
    __shared__ float Wl[64 * 65];   // +1 pad: lane halves hit distinct banks
    const int tid = threadIdx.x;
    for (int i = tid; i < 64 * 64; i += 256) {
        Wl[(i >> 6) * 65 + (i & 63)] = W[i];
    }
    __syncthreads();

    const int wid  = tid >> 5;
    const int lane = tid & 31;
    const int half = lane >> 4;        // 0: K pair {0,1}, 1: K pair {2,3}
    const int l16  = lane & 15;
    const long rowBase = (long)blockIdx.x * 128 + wid * 16;
    const int ka = half * 2;

    const float* Arow = X + (rowBase + l16) * DD;   // this lane's A row

    v8f c[4];
    if (accum) {
        #pragma unroll
        for (int nt = 0; nt < 4; ++nt) {
            const int col = nt * 16 + l16;
            #pragma unroll
            for (int i = 0; i < 8; ++i)
                c[nt][i] = Y[(rowBase + half * 8 + i) * DD + col];
        }
    } else {
        #pragma unroll
        for (int nt = 0; nt < 4; ++nt) c[nt] = (v8f)(0.0f);
    }

    #pragma unroll
    for (int ks = 0; ks < 16; ++ks) {
        const int k = ks * 4 + ka;
        v2f a = *(const v2f*)(Arow + k);                 // A[m][k], A[m][k+1]
        #pragma unroll
        for (int nt = 0; nt < 4; ++nt) {
            const int col = nt * 16 + l16;
            v2f b;
            b.x = Wl[k * 65 + col];                      // B[k][n]
            b.y = Wl[(k + 1) * 65 + col];                // B[k+1][n]
            c[nt] = __builtin_amdgcn_wmma_f32_16x16x4_f32(
                        false, a, false, b, (short)0, c[nt], false, false);
        }
    }

    #pragma unroll
    for (int nt = 0; nt < 4; ++nt) {
        const int col = nt * 16 + l16;
        float bv = has_bias ? bias[col] : 0.0f;
        #pragma unroll
        for (int i = 0; i < 8; ++i) {
            float v = c[nt][i] + bv;
            if (do_relu) v = fmaxf(v, 0.0f);
            Y[(rowBase + half * 8 + i) * DD + col] = v;
        }
    }
}

// in-place: buf = relu(buf + b_conv)
__global__ void k_bias_relu(float* __restrict__ buf, const float* __restrict__ bc) {
    long i = (long)blockIdx.x * blockDim.x + threadIdx.x;
    if (i >= (long)NEL) return;
    buf[i] = fmaxf(buf[i] + bc[i & 63], 0.0f);
}

// h += sigmoid(z + b_gate) * act
__global__ void k_combine(float* __restrict__ h, const float* __restrict__ z,
                          const float* __restrict__ act, const float* __restrict__ bg) {
    long i = (long)blockIdx.x * blockDim.x + threadIdx.x;
    if (i >= (long)NEL) return;
    float x = z[i] + bg[i & 63];
    float g = 1.0f / (1.0f + expf(-x));
    h[i] += g * act[i];
}

// graph_emb[b,s,:] = sum_n mask[b,s,n]*h[b,n,:] / max(sum_n mask,1)
__global__ void k_pool(const float* __restrict__ mask, const float* __restrict__ h,
                       float* __restrict__ out) {
    __shared__ float part[4][64];
    __shared__ float cpart[4];
    const int bs = blockIdx.x;          // b*S + s
    const int b  = bs / SS;
    const int tid = threadIdx.x;
    const int d = tid & 63;
    const int g = tid >> 6;             // 0..3
    const float* mrow = mask + (size_t)bs * NN;
    float acc = 0.0f, cnt = 0.0f;
    for (int n = g; n < NN; n += 4) {
        float m = mrow[n];
        acc += m * h[((size_t)b * NN + n) * DD + d];
        cnt += m;
    }
    part[g][d] = acc;
    if (d == 0) cpart[g] = cnt;
    __syncthreads();
    if (g == 0) {
        float s = part[0][d] + part[1][d] + part[2][d] + part[3][d];
        float ct = cpart[0] + cpart[1] + cpart[2] + cpart[3];
        out[(size_t)bs * DD + d] = s / fmaxf(ct, 1.0f);
    }
}

// ---------------- host orchestration ----------------
extern "C" void kernel_launch(void* const* d_in, const int* in_sizes, int n_in,
                              void* d_out, int out_size, void* d_ws, size_t ws_size,
                              hipStream_t stream) {
    const int*   batch_nodes = (const int*)  d_in[0];
    const int*   adj         = (const int*)  d_in[1];
    const float* mask        = (const float*)d_in[2];
    // d_in[3] = num_subgraphs (scalar) — fixed S=16
    const float* type_table  = (const float*)d_in[4];
    const float* depth_table = (const float*)d_in[5];
    const float* W_rel       = (const float*)d_in[6];   // (3,8,64,64)
    const float* b_conv      = (const float*)d_in[7];   // (3,64)
    const float* W_gate      = (const float*)d_in[8];   // (3,64,64)
    const float* b_gate      = (const float*)d_in[9];   // (3,64)
    const float* W_out       = (const float*)d_in[10];  // (2,64,64)
    const float* b_out       = (const float*)d_in[11];  // (2,64)
    float* out = (float*)d_out;

    const int E = in_sizes[1] / 5;

    // workspace layout: depths + 3 node-feature buffers (~50.6 MB total)
    int*   depths = (int*)d_ws;                 // BN ints
    float* base   = (float*)d_ws + BN;
    float* H   = base + 0 * NEL;                // node features (residual stream)
    float* SEG = base + 1 * NEL;                // typed segment sums / gate z / MLP tmp
    float* AGG = base + 2 * NEL;                // agg -> act (in-place) / MLP out

    const int ZB   = (int)((NEL + 255) / 256);      // elementwise grid over BN*64
    const int EB1  = (E + 255) / 256;               // 1 thread / edge
    const long es  = (long)E * 16;
    const int EB16 = (int)((es + 255) / 256);       // 16 threads / edge
    const int MMB  = BN / 128;                      // 512 matmul blocks

    // --- depths + embedding ---
    k_zero_f32<<<(BN + 255) / 256, 256, 0, stream>>>((float*)depths, BN);
    k_scatter_depth<<<EB1, 256, 0, stream>>>(adj, E, depths);
    k_embed<<<ZB, 256, 0, stream>>>(batch_nodes, depths, type_table, depth_table, H);

    // --- GCN layers ---
    for (int l = 0; l < LL; ++l) {
        for (int t = 0; t < NT; ++t) {
            k_zero_f32<<<ZB, 256, 0, stream>>>(SEG, (long)NEL);
            k_scatter_type<<<EB16, 256, 0, stream>>>(adj, E, t, H, SEG);
            k_matmul64<<<MMB, 256, 0, stream>>>(SEG, W_rel + ((size_t)(l * NT + t)) * 4096,
                                                nullptr, AGG, /*accum=*/(t > 0), 0, 0);
        }
        k_bias_relu<<<ZB, 256, 0, stream>>>(AGG, b_conv + l * 64);          // AGG = act
        k_matmul64<<<MMB, 256, 0, stream>>>(AGG, W_gate + (size_t)l * 4096,
                                            nullptr, SEG, 0, 0, 0);         // SEG = z
        k_combine<<<ZB, 256, 0, stream>>>(H, SEG, AGG, b_gate + l * 64);
    }

    // --- output MLP ---
    k_matmul64<<<MMB, 256, 0, stream>>>(H, W_out, b_out, SEG, 0, 1, 1);             // relu
    k_matmul64<<<MMB, 256, 0, stream>>>(SEG, W_out + 4096, b_out + 64, AGG, 0, 1, 0);

    // --- subgraph pooling ---
    k_pool<<<BB * SS, 256, 0, stream>>>(mask, AGG, out);
}